// NSE_layer_59665685676147
// MI455X (gfx1250) — compile-verified
//
#include <hip/hip_runtime.h>
#include <cstdint>

// Problem constants (fixed by the reference)
#define NXC 2048
#define NYC 1024
static constexpr int   NPTS  = NXC * NYC;
static constexpr float H_    = 6.0f / 2048.0f;      // exact: 3 * 2^-10
static constexpr float INVH  = 1.0f / H_;
static constexpr float NU_   = 0.05f;
static constexpr float NUH2  = NU_ * INVH * INVH;   // NU / H^2

#define GAS __attribute__((address_space(1)))
#define LAS __attribute__((address_space(3)))

#if defined(__has_builtin)
#  if __has_builtin(__builtin_amdgcn_global_load_async_to_lds_b32)
#    define HAVE_ASYNC_LDS 1
#  else
#    define HAVE_ASYNC_LDS 0
#  endif
#  if __has_builtin(__builtin_amdgcn_wmma_f32_16x16x4_f32)
#    define HAVE_WMMA_F32X4 1
#  else
#    define HAVE_WMMA_F32X4 0
#  endif
#else
#  define HAVE_ASYNC_LDS 0
#  define HAVE_WMMA_F32X4 0
#endif

typedef float v2f __attribute__((ext_vector_type(2)));
typedef float v8f __attribute__((ext_vector_type(8)));

// ---------------------------------------------------------------------------
// Pass 1: flow_bc (3 channels) + eq_mask, elementwise.
// out layout: [0]=loss, [1 .. 1+3N)=flow_bc, [1+3N .. 1+4N)=eq_mask
// ---------------------------------------------------------------------------
__global__ __launch_bounds__(256) void nse_bc_mask_kernel(
    const int* __restrict__ layout, const float* __restrict__ flow,
    float* __restrict__ out) {
  int idx = blockIdx.x * blockDim.x + threadIdx.x;
  if (idx >= NPTS) return;
  int b = layout[NPTS + idx];                // channel 1 of layout
  float u = flow[idx];
  float v = flow[NPTS + idx];
  float p = flow[2 * NPTS + idx];
  float ubc = (b == 2) ? 0.0f : ((b == 1) ? 3.0f : u);
  float vbc = (b == 1 || b == 2) ? 0.0f : v;
  float pbc = (b == 2 || b == 3) ? 0.0f : p;
  bool  m   = (b == 0) || (b >= 4 && b <= 11);
  out[1 + idx]            = ubc;
  out[1 + NPTS + idx]     = vbc;
  out[1 + 2 * NPTS + idx] = pbc;
  out[1 + 3 * NPTS + idx] = m ? (float)b : 0.0f;
}

// ---------------------------------------------------------------------------
// Pass 2: stencil residuals on reflect-padded flow_bc; per-block partial sums.
// Tile: 32x8 points, halo tile 10x34 per channel staged in LDS via
// GLOBAL_LOAD_ASYNC_TO_LDS_B32 (ASYNCcnt path) when available.
// ---------------------------------------------------------------------------
__global__ __launch_bounds__(256) void nse_resid_kernel(
    const int* __restrict__ layout, const float* __restrict__ out,
    float* __restrict__ partial) {
  __shared__ float tU[10][34];
  __shared__ float tV[10][34];
  __shared__ float tP[10][34];
  __shared__ float red[256];

  const int tx  = threadIdx.x;          // 0..31 (one wave per tile row)
  const int ty  = threadIdx.y;          // 0..7
  const int tid = ty * 32 + tx;
  const int bx  = blockIdx.x, by = blockIdx.y;

  const float* fu = out + 1;
  const float* fv = out + 1 + NPTS;
  const float* fp = out + 1 + 2 * NPTS;

  // Stage (10 x 34) halo tile for u,v,p with reflect boundary remap.
  for (int e = tid; e < 10 * 34; e += 256) {
    int ly = e / 34, lx = e - ly * 34;
    int sr = by * 8 + ly - 1;            // padded row -> source row (reflect)
    if (sr < 0)    sr = 1;
    if (sr >= NYC) sr = NYC - 2;
    int sc = bx * 32 + lx - 1;
    if (sc < 0)    sc = 1;
    if (sc >= NXC) sc = NXC - 2;
    int g = sr * NXC + sc;
#if HAVE_ASYNC_LDS
    __builtin_amdgcn_global_load_async_to_lds_b32(
        (GAS int*)(void*)(fu + g), (LAS int*)(void*)&tU[ly][lx], 0, 0);
    __builtin_amdgcn_global_load_async_to_lds_b32(
        (GAS int*)(void*)(fv + g), (LAS int*)(void*)&tV[ly][lx], 0, 0);
    __builtin_amdgcn_global_load_async_to_lds_b32(
        (GAS int*)(void*)(fp + g), (LAS int*)(void*)&tP[ly][lx], 0, 0);
#else
    tU[ly][lx] = fu[g];
    tV[ly][lx] = fv[g];
    tP[ly][lx] = fp[g];
#endif
  }
#if HAVE_ASYNC_LDS
#  if __has_builtin(__builtin_amdgcn_s_wait_asynccnt)
  __builtin_amdgcn_s_wait_asynccnt(0);
#  else
  asm volatile("s_wait_asynccnt 0x0" ::: "memory");
#  endif
#endif
  __syncthreads();

  const int i = bx * 32 + tx;
  const int j = by * 8 + ty;
  const int b = layout[NPTS + j * NXC + i];

  // Center values (== flow_bc at (j,i))
  const float u  = tU[ty + 1][tx + 1];
  const float v  = tV[ty + 1][tx + 1];
  const float pc = tP[ty + 1][tx + 1];

  const float uxm = tU[ty + 1][tx],     uxp = tU[ty + 1][tx + 2];
  const float uym = tU[ty][tx + 1],     uyp = tU[ty + 2][tx + 1];
  const float vxm = tV[ty + 1][tx],     vxp = tV[ty + 1][tx + 2];
  const float vym = tV[ty][tx + 1],     vyp = tV[ty + 2][tx + 1];
  const float pxm = tP[ty + 1][tx],     pxp = tP[ty + 1][tx + 2];
  const float pym = tP[ty][tx + 1],     pyp = tP[ty + 2][tx + 1];

  const float dxu = 0.5f * (uxp - uxm);
  const float dyu = 0.5f * (uyp - uym);         // Dy = 0.5*(a[2:] - a[:-2])
  const float dxv = 0.5f * (vxp - vxm);
  const float dyv = 0.5f * (vyp - vym);
  const float lapu = uym + uyp + uxm + uxp - 4.0f * u;
  const float lapv = vym + vyp + vxm + vxp - 4.0f * v;

  // Pressure gradients, one-sided where b selects (note FDy uses row j-1)
  const float dpx = (b == 4 || b == 8 || b == 11) ? (pxp - pc)
                  : (b == 6 || b == 9 || b == 10) ? (pc - pxm)
                  : 0.5f * (pxp - pxm);
  const float dpy = (b == 7 || b == 10 || b == 11) ? (pym - pc)   // FDy
                  : (b == 5 || b == 8 || b == 9)   ? (pc - pyp)   // BDy
                  : 0.5f * (pyp - pym);

  const float momu = (u * dxu + v * dyu + dpx) * INVH - NUH2 * lapu;
  const float momv = (u * dxv + v * dyv + dpy) * INVH - NUH2 * lapv;
  const float cont = (dxu + dyv) * INVH;

  const bool  m   = (b == 0) || (b >= 4 && b <= 11);
  const float loc = m ? (momu * momu + momv * momv + cont * cont) : 0.0f;

  // Deterministic block reduction
  red[tid] = loc;
  __syncthreads();
  for (int s = 128; s > 0; s >>= 1) {
    if (tid < s) red[tid] += red[tid + s];
    __syncthreads();
  }
  if (tid == 0) partial[by * gridDim.x + bx] = red[0];
}

// ---------------------------------------------------------------------------
// Pass 3: reduce per-block partials; wave-level sum via V_WMMA_F32_16X16X4_F32
// row-sum trick (B = ones, exact in f32), then cross-wave LDS combine.
// ---------------------------------------------------------------------------
__global__ __launch_bounds__(256) void nse_finalize_kernel(
    const float* __restrict__ partial, int n, float* __restrict__ out) {
  __shared__ float acc[8];
  const int tid = threadIdx.x;
  float t = 0.0f;
  for (int k = tid; k < n; k += 256) t += partial[k];

#if HAVE_WMMA_F32X4
  // A is 16x4 f32 (2 VGPRs/lane): put t in slot 0 of each lane -> the 64
  // A-slots cover all 32 lanes exactly once. D = A x ones(4x16): every D
  // element in row m equals sum_k A[m,k]; summing this lane's 8 D VGPRs gives
  // rows 0..7 (lanes 0-15) / rows 8..15 (lanes 16-31).
  v2f a;  a.x = t;     a.y = 0.0f;
  v2f bn; bn.x = 1.0f; bn.y = 1.0f;
  v8f c = {};
  v8f d = __builtin_amdgcn_wmma_f32_16x16x4_f32(
      /*neg_a=*/false, a, /*neg_b=*/false, bn,
      /*c_mod=*/(short)0, c, /*reuse_a=*/false, /*reuse_b=*/false);
  float s = d[0] + d[1] + d[2] + d[3] + d[4] + d[5] + d[6] + d[7];
  float wsum = __shfl(s, 0, 32) + __shfl(s, 16, 32);
#else
  float wsum = t;
  for (int off = 16; off > 0; off >>= 1) wsum += __shfl_down(wsum, off, 32);
  wsum = __shfl(wsum, 0, 32);
#endif

  const int lane = tid & 31, wave = tid >> 5;
  if (lane == 0) acc[wave] = wsum;
  __syncthreads();
  if (tid == 0) {
    float tot = 0.0f;
    for (int w = 0; w < 8; ++w) tot += acc[w];
    out[0] = tot / (float)NPTS;   // mean(u^2)+mean(v^2)+mean(c^2), same N
  }
}

// ---------------------------------------------------------------------------
extern "C" void kernel_launch(void* const* d_in, const int* in_sizes, int n_in,
                              void* d_out, int out_size, void* d_ws, size_t ws_size,
                              hipStream_t stream) {
  const int*   layout = (const int*)d_in[0];     // (1,2,NY,NX) int32
  const float* flow   = (const float*)d_in[1];   // (1,3,NY,NX) f32
  float* out     = (float*)d_out;                // [loss | flow_bc | eq_mask]
  float* partial = (float*)d_ws;                 // 8192 floats scratch

  nse_bc_mask_kernel<<<dim3((NPTS + 255) / 256), dim3(256), 0, stream>>>(
      layout, flow, out);

  dim3 b2(32, 8), g2(NXC / 32, NYC / 8);         // 64 x 128 = 8192 blocks
  nse_resid_kernel<<<g2, b2, 0, stream>>>(layout, out, partial);

  const int nparts = (NXC / 32) * (NYC / 8);
  nse_finalize_kernel<<<dim3(1), dim3(256), 0, stream>>>(partial, nparts, out);
}